// DownConv_12790412607762
// MI455X (gfx1250) — compile-verified
//
#include <hip/hip_runtime.h>
#include <math.h>

typedef float v2f __attribute__((ext_vector_type(2)));
typedef float v8f __attribute__((ext_vector_type(8)));

#define IN_EPS 1e-5f

// ---------------------------------------------------------------------------
// Mesh conv layer: y[b,o,e] = bias[o] + sum_{c,s} Gm[b,c,e,s] * w[o,c,s]
// One block = 16 edges, 256 threads (8 waves). Wave w computes output rows
// [16w, 16w+16) x 16 edges via V_WMMA_F32_16X16X4_F32 over K = Cin*5.
// CIN_T: compile-time Cin (64 / 128) so the K-loop has a constant trip count
// and can be unrolled; CIN_T==0 falls back to runtime Cin.
// ---------------------------------------------------------------------------
template <int CIN_T>
__global__ __launch_bounds__(256)
void meshconv_wmma(const float* __restrict__ x,    // [B, Cin, E]
                   const int*   __restrict__ gemm, // [B, E, 4]
                   const float* __restrict__ w,    // [Cout, Cin, 5]
                   const float* __restrict__ bias, // [Cout]
                   float*       __restrict__ y,    // [B, Cout, E]
                   int CinRt, int Cout, int E)
{
    const int Cin = CIN_T ? CIN_T : CinRt;
    const int K   = Cin * 5;

    extern __shared__ float smem[];                // [Cin*5][16]
    const int b  = blockIdx.y;
    const int e0 = blockIdx.x * 16;
    const int t  = threadIdx.x;

    // ---- gather phase: build Gm tile [K x 16] in LDS -----------------------
    {
        const int e  = t & 15;                     // edge within tile
        const int c0 = t >> 4;                     // 0..15 channel slice
        const int eg = e0 + e;
        const int4 g = *(const int4*)(gemm + ((size_t)b * E + eg) * 4);
        for (int c = c0; c < Cin; c += 16) {
            const float* xr = x + ((size_t)b * Cin + c) * E;
            float f0 = xr[eg];
            float f1 = (g.x >= 0) ? xr[g.x] : 0.0f;
            float f2 = (g.y >= 0) ? xr[g.y] : 0.0f;
            float f3 = (g.z >= 0) ? xr[g.z] : 0.0f;
            float f4 = (g.w >= 0) ? xr[g.w] : 0.0f;
            float* row = smem + (size_t)(c * 5) * 16 + e;
            row[0 * 16] = f0;
            row[1 * 16] = f1 + f3;
            row[2 * 16] = f2 + f4;
            row[3 * 16] = fabsf(f1 - f3);
            row[4 * 16] = fabsf(f2 - f4);
        }
    }
    __syncthreads();

    // ---- WMMA phase: D(16o x 16e) = W(16o x K) * Gm(K x 16e) ---------------
    const int lane = t & 31;
    const int wave = t >> 5;                       // 0..7
    const int half = lane >> 4;                    // 0 | 1
    const int m    = lane & 15;                    // row (A) / col (B)
    const int o0   = wave * 16;
    const float* wrow = w + (size_t)(o0 + m) * K;

    v8f acc = {};
    #pragma unroll 8
    for (int k0 = 0; k0 < K; k0 += 4) {
        // ISA 16x4 f32 A layout: lanes 0-15 hold K=k0,k0+1; lanes 16-31 K=k0+2,k0+3
        const int kk = k0 + 2 * half;
        v2f a, bm;
        a.x  = wrow[kk];
        a.y  = wrow[kk + 1];
        bm.x = smem[kk * 16 + m];
        bm.y = smem[(kk + 1) * 16 + m];
        acc = __builtin_amdgcn_wmma_f32_16x16x4_f32(false, a, false, bm,
                                                    (short)0, acc, false, false);
    }

    // ---- store: C/D layout: VGPR j -> row o0+j (lanes 0-15) / o0+j+8 -------
    const int e = e0 + m;
    #pragma unroll
    for (int j = 0; j < 8; ++j) {
        const int o = o0 + j + 8 * half;
        y[((size_t)b * Cout + o) * E + e] = acc[j] + bias[o];
    }
}

// ---------------------------------------------------------------------------
// Instance-norm statistics: one block per (b,c); coalesced b128 pass over E.
// ---------------------------------------------------------------------------
__global__ __launch_bounds__(256)
void inorm_stats(const float* __restrict__ y, float* __restrict__ mean,
                 float* __restrict__ rstd, int E)
{
    __shared__ float s1[256];
    __shared__ float s2[256];
    const int bc = blockIdx.x;
    const float* row = y + (size_t)bc * E;
    const int E4 = E >> 2;                          // float4 chunks

    float a = 0.0f, b = 0.0f;
    for (int i = threadIdx.x; i < E4; i += 256) {
        float4 v = ((const float4*)row)[i];
        a += v.x + v.y + v.z + v.w;
        b += v.x * v.x + v.y * v.y + v.z * v.z + v.w * v.w;
    }
    for (int i = (E4 << 2) + threadIdx.x; i < E; i += 256) {   // tail
        float v = row[i];
        a += v;
        b += v * v;
    }
    s1[threadIdx.x] = a;
    s2[threadIdx.x] = b;
    __syncthreads();
    for (int off = 128; off > 0; off >>= 1) {
        if (threadIdx.x < off) {
            s1[threadIdx.x] += s1[threadIdx.x + off];
            s2[threadIdx.x] += s2[threadIdx.x + off];
        }
        __syncthreads();
    }
    if (threadIdx.x == 0) {
        float inv = 1.0f / (float)E;
        float mu  = s1[0] * inv;
        float var = s2[0] * inv - mu * mu;
        mean[bc] = mu;
        rstd[bc] = rsqrtf(var + IN_EPS);
    }
}

// ---------------------------------------------------------------------------
// x1 = relu(instance_norm(y)); 2-D grid: blockIdx.y = (b,c) row, float4 lanes.
// ---------------------------------------------------------------------------
__global__ __launch_bounds__(256)
void norm_relu(const float* __restrict__ y, const float* __restrict__ mean,
               const float* __restrict__ rstd, float* __restrict__ out, int E)
{
    const int  bc   = blockIdx.y;
    const int  i4   = (blockIdx.x * 256 + threadIdx.x) * 4;
    if (i4 >= E) return;
    const size_t base = (size_t)bc * E + i4;
    const float  mu = mean[bc], rs = rstd[bc];
    if (i4 + 3 < E) {
        float4 v = *(const float4*)(y + base);
        float4 r;
        r.x = fmaxf((v.x - mu) * rs, 0.0f);
        r.y = fmaxf((v.y - mu) * rs, 0.0f);
        r.z = fmaxf((v.z - mu) * rs, 0.0f);
        r.w = fmaxf((v.w - mu) * rs, 0.0f);
        *(float4*)(out + base) = r;
    } else {
        for (int j = 0; i4 + j < E; ++j)
            out[base + j] = fmaxf((y[base + j] - mu) * rs, 0.0f);
    }
}

// out = relu(instance_norm(y2) + x1)
__global__ __launch_bounds__(256)
void norm_res_relu(const float* __restrict__ y, const float* __restrict__ mean,
                   const float* __restrict__ rstd, const float* __restrict__ res,
                   float* __restrict__ out, int E)
{
    const int  bc   = blockIdx.y;
    const int  i4   = (blockIdx.x * 256 + threadIdx.x) * 4;
    if (i4 >= E) return;
    const size_t base = (size_t)bc * E + i4;
    const float  mu = mean[bc], rs = rstd[bc];
    if (i4 + 3 < E) {
        float4 v = *(const float4*)(y + base);
        float4 x = *(const float4*)(res + base);
        float4 r;
        r.x = fmaxf((v.x - mu) * rs + x.x, 0.0f);
        r.y = fmaxf((v.y - mu) * rs + x.y, 0.0f);
        r.z = fmaxf((v.z - mu) * rs + x.z, 0.0f);
        r.w = fmaxf((v.w - mu) * rs + x.w, 0.0f);
        *(float4*)(out + base) = r;
    } else {
        for (int j = 0; i4 + j < E; ++j)
            out[base + j] = fmaxf((y[base + j] - mu) * rs + res[base + j], 0.0f);
    }
}

// ---------------------------------------------------------------------------
static inline void launch_meshconv(const float* x, const int* gemm, const float* w,
                                   const float* bias, float* y,
                                   int Cin, int Cout, int E, int B,
                                   hipStream_t stream)
{
    const dim3 blk(256);
    const dim3 grid((E + 15) / 16, B);
    const size_t lds = (size_t)(Cin * 5) * 16 * sizeof(float);
    if (Cin == 64)
        meshconv_wmma<64><<<grid, blk, lds, stream>>>(x, gemm, w, bias, y, Cin, Cout, E);
    else if (Cin == 128)
        meshconv_wmma<128><<<grid, blk, lds, stream>>>(x, gemm, w, bias, y, Cin, Cout, E);
    else
        meshconv_wmma<0><<<grid, blk, lds, stream>>>(x, gemm, w, bias, y, Cin, Cout, E);
}

extern "C" void kernel_launch(void* const* d_in, const int* in_sizes, int n_in,
                              void* d_out, int out_size, void* d_ws, size_t ws_size,
                              hipStream_t stream)
{
    const float* fe   = (const float*)d_in[0];
    const int*   gemm = (const int*)  d_in[1];
    const float* w1   = (const float*)d_in[2];
    const float* b1   = (const float*)d_in[3];
    const float* w2   = (const float*)d_in[4];
    const float* b2   = (const float*)d_in[5];

    const int B    = 2;
    const int Cout = in_sizes[3];                    // 128
    const int Cin  = in_sizes[2] / (Cout * 5);       // 64
    const int E    = in_sizes[0] / (B * Cin);        // 100000

    const long chanN = (long)B * Cout * E;           // 25.6M elements

    float* ybuf = (float*)d_ws;                      // [B, Cout, E]
    float* x1   = ybuf + chanN;                      // [B, Cout, E]
    float* mean = x1 + chanN;                        // [B*Cout]
    float* rstd = mean + (size_t)B * Cout;           // [B*Cout]

    const dim3 blk(256);
    const int  statsGrid = B * Cout;
    const dim3 ewGrid((E + 4 * 256 - 1) / (4 * 256), B * Cout);

    // ---- layer 1: mesh_conv -> instance_norm -> relu ----
    launch_meshconv(fe, gemm, w1, b1, ybuf, Cin, Cout, E, B, stream);
    inorm_stats<<<statsGrid, blk, 0, stream>>>(ybuf, mean, rstd, E);
    norm_relu<<<ewGrid, blk, 0, stream>>>(ybuf, mean, rstd, x1, E);

    // ---- layer 2: mesh_conv -> instance_norm -> +x1 -> relu ----
    launch_meshconv(x1, gemm, w2, b2, ybuf, Cout, Cout, E, B, stream);
    inorm_stats<<<statsGrid, blk, 0, stream>>>(ybuf, mean, rstd, E);
    norm_res_relu<<<ewGrid, blk, 0, stream>>>(ybuf, mean, rstd, x1, (float*)d_out, E);
}